// UnknownRegionDTLoss_82832739271058
// MI455X (gfx1250) — compile-verified
//
#include <hip/hip_runtime.h>
#include <math.h>

#define B_ 4
#define C_ 3
#define H_ 384
#define W_ 384

typedef __attribute__((ext_vector_type(2))) float v2f;
typedef __attribute__((ext_vector_type(8))) float v8f;

// acc layout in d_ws: acc[0..3] = per-image weighted sums, acc[4..7] = per-image
// foreground counts, acc[8..15] pad.  gT (transposed squared row-distances,
// [B][W][H]) follows at acc+16.

// One block per (b,w) column; thread = output row h.
// CDNA5 paths: async global->LDS staging of the column, global_prefetch for the
// strided per-pixel loads, (min,+) parabola pass from LDS broadcasts, and a
// WMMA_F32_16X16X4_F32 matrix-pipe reduction (total = 1^T * A * 1) for the
// per-block sums.
__global__ void edt_colpass_kernel(const float* __restrict__ logits,
                                   const int* __restrict__ target,
                                   const float* __restrict__ gT,
                                   float* __restrict__ acc) {
    __shared__ float s_g[H_];
    __shared__ float s_red[H_];
    __shared__ float s_cnt[H_];
    const int b = blockIdx.x / W_;
    const int w = blockIdx.x % W_;
    const int h = threadIdx.x;

    // Async copy: each lane pulls one float of the column straight into LDS.
    {
        unsigned lds_off = (unsigned)(size_t)(&s_g[h]);   // low 32 bits = LDS addr
        const float* src = gT + ((size_t)(b * W_ + w) * H_ + h);
        asm volatile("global_load_async_to_lds_b32 %0, %1, off"
                     :: "v"(lds_off), "v"(src)
                     : "memory");
    }

    // Prefetch the strided pixel data needed after the long loop.
    const size_t plane = (size_t)H_ * W_;
    const size_t pix   = (size_t)(b * H_ + h) * W_ + w;
    const size_t lbase = (size_t)b * C_ * plane + (size_t)h * W_ + w;
    __builtin_prefetch(&target[pix], 0, 0);
    __builtin_prefetch(&logits[lbase], 0, 0);
    __builtin_prefetch(&logits[lbase + plane], 0, 0);
    __builtin_prefetch(&logits[lbase + 2 * plane], 0, 0);

    asm volatile("s_wait_asynccnt 0" ::: "memory");
    __syncthreads();

    // dsq[h] = min_j ( g[j] + (h-j)^2 )  -- (min,+) semiring, VALU + LDS broadcast.
    float dsq = 3.0e38f;
#pragma unroll 8
    for (int j = 0; j < H_; ++j) {
        float d = (float)(h - j);
        dsq = fminf(dsq, fmaf(d, d, s_g[j]));
    }

    float contrib = 0.0f, cnt = 0.0f;
    if (target[pix] == 1) {
        float l0 = logits[lbase];
        float l1 = logits[lbase + plane];
        float l2 = logits[lbase + 2 * plane];
        float m  = fmaxf(l0, fmaxf(l1, l2));
        float e0 = expf(l0 - m), e1 = expf(l1 - m), e2 = expf(l2 - m);
        float p1 = e1 / (e0 + e1 + e2);
        contrib = (1.0f - p1) * sqrtf(dsq);
        cnt = 1.0f;
    }

    s_red[h] = contrib;
    s_cnt[h] = cnt;
    __syncthreads();

    // Matrix-pipe block reduction: wave 0 (full EXEC, all 32 lanes) sums the 384
    // partials as 6 chunks of 64 via D = A x ones(4x16) + C, fp32 throughout.
    if (h < 32) {
        v8f accw = {};
        v8f accc = {};
        v2f ones; ones.x = 1.0f; ones.y = 1.0f;
        for (int c = 0; c < H_; c += 64) {
            v2f a, k;
            a.x = s_red[c + 2 * h];
            a.y = s_red[c + 2 * h + 1];
            k.x = s_cnt[c + 2 * h];
            k.y = s_cnt[c + 2 * h + 1];
            accw = __builtin_amdgcn_wmma_f32_16x16x4_f32(
                false, a, false, ones, (short)0, accw, false, false);
            accc = __builtin_amdgcn_wmma_f32_16x16x4_f32(
                false, k, false, ones, (short)0, accc, false, false);
        }
        // Column 0 of D: lane 0 holds M=0..7, lane 16 holds M=8..15.
        float lw = accw[0] + accw[1] + accw[2] + accw[3] +
                   accw[4] + accw[5] + accw[6] + accw[7];
        float lc = accc[0] + accc[1] + accc[2] + accc[3] +
                   accc[4] + accc[5] + accc[6] + accc[7];
        float tw = __shfl(lw, 0, 32) + __shfl(lw, 16, 32);
        float tc = __shfl(lc, 0, 32) + __shfl(lc, 16, 32);
        if (h == 0) {
            atomicAdd(&acc[b],     tw);
            atomicAdd(&acc[4 + b], tc);
        }
    }
}

__global__ void edt_init_kernel(float* __restrict__ acc) {
    if (threadIdx.x < 16) acc[threadIdx.x] = 0.0f;
}

// One block per (b,h) row; 384 threads; LDS Hillis-Steele max/min scans give the
// nearest-zero index left/right of each pixel -> 1-D distance -> squared, stored
// transposed so the column pass reads contiguously.
__global__ void edt_rowpass_kernel(const int* __restrict__ target,
                                   float* __restrict__ gT) {
    __shared__ int s_left[W_];
    __shared__ int s_right[W_];
    const int b = blockIdx.x / H_;
    const int h = blockIdx.x % H_;
    const int t = threadIdx.x;

    const int tgt = target[(b * H_ + h) * W_ + t];
    const bool iszero = (tgt != 1);
    s_left[t]  = iszero ? t : -10000000;
    s_right[t] = iszero ? t :  10000000;
    __syncthreads();

    for (int off = 1; off < W_; off <<= 1) {
        int vl = (t >= off)     ? s_left[t - off]  : -10000000;
        int vr = (t + off < W_) ? s_right[t + off] :  10000000;
        __syncthreads();
        s_left[t]  = max(s_left[t], vl);
        s_right[t] = min(s_right[t], vr);
        __syncthreads();
    }

    float fl = (float)(t - s_left[t]);
    float fr = (float)(s_right[t] - t);
    float d1 = fminf(fminf(fl, fr), 1.0e7f);   // BIG clamp, matches reference
    gT[((size_t)(b * W_ + t)) * H_ + h] = d1 * d1;
}

__global__ void edt_final_kernel(const float* __restrict__ acc,
                                 float* __restrict__ out) {
    if (threadIdx.x == 0) {
        const float inv = 1.0f / (float)(H_ * W_);
        float s = 0.0f;
        for (int b = 0; b < B_; ++b) {
            float per = (acc[4 + b] > 0.0f) ? acc[b] * inv : 0.0f;
            s += per;
        }
        out[0] = s * 0.25f;
    }
}

extern "C" void kernel_launch(void* const* d_in, const int* in_sizes, int n_in,
                              void* d_out, int out_size, void* d_ws, size_t ws_size,
                              hipStream_t stream) {
    (void)in_sizes; (void)n_in; (void)out_size; (void)ws_size;
    const float* logits = (const float*)d_in[0];   // [4,3,384,384] f32
    const int*   target = (const int*)d_in[1];     // [4,384,384] int
    float* acc = (float*)d_ws;                     // 16 floats
    float* gT  = acc + 16;                         // [4,384,384] f32 transposed

    edt_init_kernel<<<1, 32, 0, stream>>>(acc);
    edt_rowpass_kernel<<<B_ * H_, W_, 0, stream>>>(target, gT);
    edt_colpass_kernel<<<B_ * W_, H_, 0, stream>>>(logits, target, gT, acc);
    edt_final_kernel<<<1, 32, 0, stream>>>(acc, (float*)d_out);
}